// CrossHeadDeltaQuantizer_46162308497802
// MI455X (gfx1250) — compile-verified
//
#include <hip/hip_runtime.h>
#include <math.h>

typedef __attribute__((ext_vector_type(2))) float v2f;
typedef __attribute__((ext_vector_type(8))) float v8f;

#define TS      130          // padded LDS row stride (floats) -> conflict-free B fetches
#define NWAVES  4
#define DDIM    128
#define SDIM    4096
#define HDIM    8
#define EPSF    1e-8f

// cross-lane XOR butterfly via ds_swizzle (group-of-32: offset[14:10]=xor, [4:0]=and)
#define SWZ_XOR(v, m) \
  __int_as_float(__builtin_amdgcn_ds_swizzle(__float_as_int(v), (((m) << 10) | 0x1f)))

__global__ __launch_bounds__(128, 1)
void chdq_kernel(const float* __restrict__ kv,
                 const float* __restrict__ Ra_g,
                 const float* __restrict__ cba,
                 const float* __restrict__ Rd_g,
                 const float* __restrict__ cbd,
                 float* __restrict__ out)
{
  extern __shared__ float smem[];
  const int tid  = threadIdx.x;
  const int wave = tid >> 5;
  const int lane = tid & 31;
  const int ln   = lane & 15;       // lane within half
  const int hi   = lane >> 4;       // which 16-lane half
  const int kh   = hi << 1;         // K sub-offset for f32 WMMA A/B layout

  float* Ra = smem;                                         // 128*TS
  float* Rd = smem + DDIM * TS;                             // 128*TS
  float* tA = smem + 2 * DDIM * TS + wave * (16 * TS);      // anchor recon (A-layout src)
  float* tB = smem + 2 * DDIM * TS + NWAVES * 16 * TS + wave * (16 * TS); // staging
  float* rv = smem + 2 * DDIM * TS + 2 * NWAVES * 16 * TS + wave * 16;   // row norms

  // ---- stage both rotation matrices into padded LDS (coalesced b128 reads) ----
  for (int i = tid * 4; i < DDIM * DDIM; i += 128 * 4) {
    const int r = i >> 7, c = i & 127;
    float4 a = *(const float4*)(Ra_g + i);
    Ra[r * TS + c + 0] = a.x; Ra[r * TS + c + 1] = a.y;
    Ra[r * TS + c + 2] = a.z; Ra[r * TS + c + 3] = a.w;
    float4 d = *(const float4*)(Rd_g + i);
    Rd[r * TS + c + 0] = d.x; Rd[r * TS + c + 1] = d.y;
    Rd[r * TS + c + 2] = d.z; Rd[r * TS + c + 3] = d.w;
  }
  __syncthreads();

  const int b  = blockIdx.x >> 6;                    // 64 blocks per batch
  const int s0 = ((blockIdx.x & 63) << 6) + (wave << 4);

  const float ca0 = cba[0], ca1 = cba[1], ca2 = cba[2], ca3 = cba[3];
  const float t01 = 0.5f * (ca0 + ca1), t12 = 0.5f * (ca1 + ca2), t23 = 0.5f * (ca2 + ca3);
  const float cd0 = cbd[0], cd1 = cbd[1];

  // ================= anchor head =================
  v2f x[32];                                         // A-layout operand (row = ln, K pairs)
  {
    const float* src = kv + ((size_t)(b * HDIM) * SDIM + s0 + ln) * DDIM + kh;
    float n2 = 0.f;
#pragma unroll
    for (int c = 0; c < 32; ++c) {
      x[c] = *(const v2f*)(src + 4 * c);
      n2 += x[c].x * x[c].x + x[c].y * x[c].y;
    }
    n2 += SWZ_XOR(n2, 16);                           // combine the two K-halves of each row
    const float an  = sqrtf(n2);
    const float inv = 1.f / (an + EPSF);
#pragma unroll
    for (int c = 0; c < 32; ++c) { x[c].x *= inv; x[c].y *= inv; }
    if (hi == 0) rv[ln] = an;
  }

  // y = x_hat @ Ra^T  (fp32 WMMA, K accumulated 4 at a time)
  v8f y[8];
#pragma unroll
  for (int t = 0; t < 8; ++t) {
    v8f acc = {};
    const float* bp = Ra + (16 * t + ln) * TS + kh;  // B[kk][nn] = Ra[n][k] -> contiguous pair
#pragma unroll
    for (int c = 0; c < 32; ++c) {
      v2f bv = *(const v2f*)(bp + 4 * c);
      acc = __builtin_amdgcn_wmma_f32_16x16x4_f32(false, x[c], false, bv,
                                                  (short)0, acc, false, false);
    }
    y[t] = acc;
  }

  // Lloyd-Max 2-bit quantize; alpha = mean|res| per row via XOR butterfly
  float sr[8] = {0.f, 0.f, 0.f, 0.f, 0.f, 0.f, 0.f, 0.f};
#pragma unroll
  for (int t = 0; t < 8; ++t)
#pragma unroll
    for (int r = 0; r < 8; ++r) {
      const float e = y[t][r];
      const float q = (e <= t12) ? ((e <= t01) ? ca0 : ca1)
                                 : ((e <= t23) ? ca2 : ca3);
      sr[r] += fabsf(e - q);
    }
  float alpha[8];
#pragma unroll
  for (int r = 0; r < 8; ++r) {
    float s = sr[r];
    s += SWZ_XOR(s, 1); s += SWZ_XOR(s, 2);
    s += SWZ_XOR(s, 4); s += SWZ_XOR(s, 8);
    alpha[r] = s * (1.0f / 128.0f);
  }
#pragma unroll
  for (int t = 0; t < 8; ++t)
#pragma unroll
    for (int r = 0; r < 8; ++r) {
      const float e = y[t][r];
      const float q = (e <= t12) ? ((e <= t01) ? ca0 : ca1)
                                 : ((e <= t23) ? ca2 : ca3);
      const float res = e - q;
      const float sgn = (res > 0.f) ? 1.f : ((res < 0.f) ? -1.f : 0.f);
      tB[(r + 8 * hi) * TS + 16 * t + ln] = q + alpha[r] * sgn;   // C->A relayout via LDS
    }
  __syncthreads();

  // recon = (y_hat @ Ra) * n
#pragma unroll
  for (int c = 0; c < 32; ++c)
    x[c] = *(const v2f*)(tB + ln * TS + 4 * c + kh);
  float nv[8];
#pragma unroll
  for (int r = 0; r < 8; ++r) nv[r] = rv[r + 8 * hi];

  v8f rec[8];
#pragma unroll
  for (int t = 0; t < 8; ++t) {
    v8f acc = {};
#pragma unroll
    for (int c = 0; c < 32; ++c) {
      const int kr = 4 * c + kh;
      v2f bv;
      bv.x = Ra[kr * TS + 16 * t + ln];              // B[kk][nn] = Ra[k][n]
      bv.y = Ra[(kr + 1) * TS + 16 * t + ln];
      acc = __builtin_amdgcn_wmma_f32_16x16x4_f32(false, x[c], false, bv,
                                                  (short)0, acc, false, false);
    }
#pragma unroll
    for (int r = 0; r < 8; ++r) acc[r] *= nv[r];
    rec[t] = acc;                                    // keep anchor recon in registers
  }

  {
    float* dst = out + ((size_t)(b * HDIM) * SDIM + s0) * DDIM;
#pragma unroll
    for (int t = 0; t < 8; ++t)
#pragma unroll
      for (int r = 0; r < 8; ++r) {
        const int m = r + 8 * hi, n = 16 * t + ln;
        dst[(size_t)m * DDIM + n] = rec[t][r];
        tA[m * TS + n] = rec[t][r];                  // A-layout source for deltas
      }
  }
  __syncthreads();

  // ================= delta heads (1-bit polar: sign is norm-invariant) =================
  for (int h = 1; h < HDIM; ++h) {
    const float* src = kv + ((size_t)(b * HDIM + h) * SDIM + s0 + ln) * DDIM + kh;
    float n2 = 0.f;
#pragma unroll
    for (int c = 0; c < 32; ++c) {
      v2f v = *(const v2f*)(src + 4 * c);
      v2f a = *(const v2f*)(tA + ln * TS + 4 * c + kh);
      v.x -= a.x; v.y -= a.y;
      x[c] = v;
      n2 += v.x * v.x + v.y * v.y;
    }
    n2 += SWZ_XOR(n2, 16);
    const float dn = sqrtf(n2);
    if (hi == 0) rv[ln] = dn;

    // z = delta @ Rd^T ; codes = (z<=0 ? cd0 : cd1), written straight to staging tile
#pragma unroll
    for (int t = 0; t < 8; ++t) {
      v8f acc = {};
      const float* bp = Rd + (16 * t + ln) * TS + kh;
#pragma unroll
      for (int c = 0; c < 32; ++c) {
        v2f bv = *(const v2f*)(bp + 4 * c);
        acc = __builtin_amdgcn_wmma_f32_16x16x4_f32(false, x[c], false, bv,
                                                    (short)0, acc, false, false);
      }
#pragma unroll
      for (int r = 0; r < 8; ++r)
        tB[(r + 8 * hi) * TS + 16 * t + ln] = (acc[r] <= 0.f) ? cd0 : cd1;
    }
    __syncthreads();

#pragma unroll
    for (int c = 0; c < 32; ++c)
      x[c] = *(const v2f*)(tB + ln * TS + 4 * c + kh);
    float dnv[8];
#pragma unroll
    for (int r = 0; r < 8; ++r) dnv[r] = rv[r + 8 * hi];

    float* dst = out + ((size_t)(b * HDIM + h) * SDIM + s0) * DDIM;
#pragma unroll
    for (int t = 0; t < 8; ++t) {
      v8f acc = {};
#pragma unroll
      for (int c = 0; c < 32; ++c) {
        const int kr = 4 * c + kh;
        v2f bv;
        bv.x = Rd[kr * TS + 16 * t + ln];
        bv.y = Rd[(kr + 1) * TS + 16 * t + ln];
        acc = __builtin_amdgcn_wmma_f32_16x16x4_f32(false, x[c], false, bv,
                                                    (short)0, acc, false, false);
      }
#pragma unroll
      for (int r = 0; r < 8; ++r)
        dst[(size_t)(r + 8 * hi) * DDIM + 16 * t + ln] = rec[t][r] + acc[r] * dnv[r];
    }
    __syncthreads();
  }
}

extern "C" void kernel_launch(void* const* d_in, const int* in_sizes, int n_in,
                              void* d_out, int out_size, void* d_ws, size_t ws_size,
                              hipStream_t stream) {
  const float* kv  = (const float*)d_in[0];   // [8,8,4096,128] f32
  const float* Ra  = (const float*)d_in[1];   // [128,128]
  const float* cba = (const float*)d_in[2];   // [4]
  const float* Rd  = (const float*)d_in[3];   // [128,128]
  const float* cbd = (const float*)d_in[4];   // [2]
  float* out = (float*)d_out;

  const size_t smem = (size_t)(2 * DDIM * TS + 2 * NWAVES * 16 * TS + NWAVES * 16)
                      * sizeof(float);        // ~195 KB of the 320 KB WGP LDS
  (void)hipFuncSetAttribute((const void*)chdq_kernel,
                            hipFuncAttributeMaxDynamicSharedMemorySize, (int)smem);

  const int blocks = 8 * (SDIM / (NWAVES * 16));  // 512
  chdq_kernel<<<blocks, 128, smem, stream>>>(kv, Ra, cba, Rd, cbd, out);
}